// MYSOCA_5600637354650
// MI455X (gfx1250) — compile-verified
//
#include <hip/hip_runtime.h>
#include <math.h>

// ---------------------------------------------------------------------------
// SOCA / MPN-COV attention for MI455X (gfx1250), fp32 throughout.
// Pass 1: Gram matrix via V_WMMA_F32_16X16X4_F32 (split-K over 1024 blocks)
// Pass 2: Newton-Schulz matrix sqrt (64x64) with the same WMMA tile routine
// Pass 3: bandwidth-bound scale pass (float4 streams)
// Workspace layout (floats): gram[32*64*64] | rowsum[32*64] | y[32*64]
// ---------------------------------------------------------------------------

typedef __attribute__((ext_vector_type(2))) float v2f;
typedef __attribute__((ext_vector_type(8))) float v8f;

#define BATCH  32
#define CDIM   64
#define MTOT   16384      // 128*128
#define SPLITK 32         // K-slices per batch in pass 1
#define KCH    64         // K staged per LDS chunk

__device__ __forceinline__ v8f wmma_f32_k4(v2f a, v2f b, v8f c) {
  // D(16x16,f32) = A(16x4,f32) * B(4x16,f32) + C
  return __builtin_amdgcn_wmma_f32_16x16x4_f32(
      /*neg_a=*/false, a, /*neg_b=*/false, b,
      /*c_mod=*/(short)0, c, /*reuse_a=*/false, /*reuse_b=*/false);
}

__device__ __forceinline__ v8f v8f_zero() {
  v8f z = {0.f, 0.f, 0.f, 0.f, 0.f, 0.f, 0.f, 0.f};
  return z;
}

// ---------------------------------------------------------------------------
__global__ __launch_bounds__(256) void soca_zero(float* __restrict__ p, int n) {
  for (int i = blockIdx.x * 256 + threadIdx.x; i < n; i += gridDim.x * 256)
    p[i] = 0.f;
}

// ---------------------------------------------------------------------------
// Pass 1: partial Gram (XdXd^T) + row sums for one (batch, k-slice).
__global__ __launch_bounds__(256) void soca_gram(const float* __restrict__ prev,
                                                 const float* __restrict__ x,
                                                 float* __restrict__ gram,
                                                 float* __restrict__ rowsum) {
  __shared__ float Xs[CDIM * KCH];                 // 16 KB: 64 rows x 64 k
  const int b    = blockIdx.x >> 5;                // /SPLITK
  const int sl   = blockIdx.x & (SPLITK - 1);
  const int tid  = threadIdx.x;
  const int lane = tid & 31;
  const int wave = tid >> 5;
  const int kslice = MTOT / SPLITK;                // 512
  const size_t base = (size_t)b * CDIM * (size_t)MTOT;

  // each wave owns tiles (wave) and (wave+8) of the 4x4 tiling of 64x64
  const int t0 = wave,      ti0 = t0 >> 2, tj0 = t0 & 3;
  const int t1 = wave + 8,  ti1 = t1 >> 2, tj1 = t1 & 3;
  const int m16 = lane & 15;
  const int khi = (lane >> 4) << 1;                // k sub-offset 0 or 2

  v8f acc0 = v8f_zero(), acc1 = v8f_zero();
  float rs = 0.f;
  const int srow = tid & 63, sseg = (tid >> 6) << 4;

  const int kbeg = sl * kslice, kend = kbeg + kslice;
  for (int k0 = kbeg; k0 < kend; k0 += KCH) {
    // cooperative staged load of xd = x - prev (coalesced float4)
#pragma unroll
    for (int i = 0; i < 4; ++i) {
      const int id  = i * 256 + tid;
      const int row = id >> 4;
      const int col = (id & 15) << 2;
      const size_t g = base + (size_t)row * MTOT + (size_t)(k0 + col);
      const float4 xv = *(const float4*)(x + g);
      const float4 pv = *(const float4*)(prev + g);
      float4 d;
      d.x = xv.x - pv.x; d.y = xv.y - pv.y;
      d.z = xv.z - pv.z; d.w = xv.w - pv.w;
      *(float4*)(&Xs[row * KCH + col]) = d;
      if (k0 + KCH < kend) {                       // prefetch next chunk
        __builtin_prefetch(x + g + KCH, 0, 0);
        __builtin_prefetch(prev + g + KCH, 0, 0);
      }
    }
    __syncthreads();

    // per-row sum partials (mean correction handled in pass 2)
#pragma unroll
    for (int j = 0; j < 16; ++j) rs += Xs[srow * KCH + sseg + j];

    // Gram accumulation: both A and B fragments are contiguous float2 rows
#pragma unroll
    for (int kk = 0; kk < KCH; kk += 4) {
      v2f a0 = *(const v2f*)(&Xs[(ti0 * 16 + m16) * KCH + kk + khi]);
      v2f b0 = *(const v2f*)(&Xs[(tj0 * 16 + m16) * KCH + kk + khi]);
      acc0 = wmma_f32_k4(a0, b0, acc0);
      v2f a1 = *(const v2f*)(&Xs[(ti1 * 16 + m16) * KCH + kk + khi]);
      v2f b1 = *(const v2f*)(&Xs[(tj1 * 16 + m16) * KCH + kk + khi]);
      acc1 = wmma_f32_k4(a1, b1, acc1);
    }
    __syncthreads();
  }

  atomicAdd(&rowsum[b * CDIM + srow], rs);
  float* gb = gram + (size_t)b * CDIM * CDIM;
#pragma unroll
  for (int r = 0; r < 8; ++r) {
    const int rr = r + ((lane >> 4) << 3);
    atomicAdd(&gb[(ti0 * 16 + rr) * CDIM + tj0 * 16 + m16], acc0[r]);
    atomicAdd(&gb[(ti1 * 16 + rr) * CDIM + tj1 * 16 + m16], acc1[r]);
  }
}

// ---------------------------------------------------------------------------
// 64x64 fp32 matmul in LDS, 8 waves x 2 tiles, WMMA f32 16x16x4.
// ns_mode==1: D = 1.5*I - 0.5*(A@B);  else D = scale*(A@B)
__device__ __forceinline__ void mm64(const float* A, const float* B, float* D,
                                     int ns_mode, float scale) {
  const int tid  = threadIdx.x;
  const int lane = tid & 31;
  const int wave = tid >> 5;
  const int m16  = lane & 15;
  const int khi  = (lane >> 4) << 1;
  for (int t = wave; t < 16; t += 8) {
    const int ti = t >> 2, tj = t & 3;
    v8f acc = v8f_zero();
#pragma unroll
    for (int kk = 0; kk < CDIM; kk += 4) {
      v2f a = *(const v2f*)(&A[(ti * 16 + m16) * CDIM + kk + khi]);
      v2f bb;
      bb.x = B[(kk + khi) * CDIM + tj * 16 + m16];
      bb.y = B[(kk + khi + 1) * CDIM + tj * 16 + m16];
      acc = wmma_f32_k4(a, bb, acc);
    }
    const int col = tj * 16 + m16;
#pragma unroll
    for (int r = 0; r < 8; ++r) {
      const int row = ti * 16 + r + ((lane >> 4) << 3);
      float v = acc[r];
      v = ns_mode ? ((row == col ? 1.5f : 0.f) - 0.5f * v) : v * scale;
      D[row * CDIM + col] = v;
    }
  }
}

// ---------------------------------------------------------------------------
// Pass 2: cov finalize, Newton-Schulz sqrtm (iterN=5), col-mean, FC, sigmoid.
__global__ __launch_bounds__(256) void soca_ns(const float* __restrict__ gram,
                                               const float* __restrict__ rowsum,
                                               const float* __restrict__ w1,
                                               const float* __restrict__ b1,
                                               const float* __restrict__ w2,
                                               const float* __restrict__ b2,
                                               float* __restrict__ yout) {
  __shared__ float lds[4 * CDIM * CDIM];           // 4 x 16 KB buffers
  const int b   = blockIdx.x;
  const int tid = threadIdx.x;
  const float invM = 1.0f / (float)MTOT;

  // cov = gram/M - mu mu^T  -> buf0
  const float* gb = gram + (size_t)b * CDIM * CDIM;
  const float* rb = rowsum + b * CDIM;
  for (int i = tid; i < CDIM * CDIM; i += 256) {
    const int r = i >> 6, c = i & 63;
    const float mu_r = rb[r] * invM, mu_c = rb[c] * invM;
    lds[i] = gb[i] * invM - mu_r * mu_c;
  }
  __syncthreads();

  // every thread reduces the diagonal from LDS (cheap, uniform result)
  float norm = 0.f;
  for (int i = 0; i < CDIM; ++i) norm += lds[i * CDIM + i];
  const float inv_norm = 1.0f / norm;
  __syncthreads();

  // An -> buf0 ; Z0 = 1.5I - 0.5*An -> buf1
  for (int i = tid; i < CDIM * CDIM; i += 256) {
    const float an = lds[i] * inv_norm;
    lds[i] = an;
    const int r = i >> 6, c = i & 63;
    lds[4096 + i] = (r == c ? 1.5f : 0.f) - 0.5f * an;
  }
  __syncthreads();

  // Y0 = An @ Z0 -> buf2
  mm64(lds, lds + 4096, lds + 2 * 4096, 0, 1.f);
  __syncthreads();

  int iY = 2, iZ = 1, iT = 3, iF = 0;
  for (int it = 0; it < 3; ++it) {                 // iterN=5 -> 3 middle iters
    mm64(lds + iZ * 4096, lds + iY * 4096, lds + iT * 4096, 1, 0.f);  // T=ns(Z@Y)
    __syncthreads();
    mm64(lds + iY * 4096, lds + iT * 4096, lds + iF * 4096, 0, 1.f);  // Y'=Y@T
    __syncthreads();
    mm64(lds + iT * 4096, lds + iZ * 4096, lds + iY * 4096, 0, 1.f);  // Z'=T@Z
    __syncthreads();
    const int oY = iY; iY = iF; iF = iZ; iZ = oY;
  }
  // final: R = sqrt(norm) * Y @ (1.5I - 0.5*(Z@Y))
  mm64(lds + iZ * 4096, lds + iY * 4096, lds + iT * 4096, 1, 0.f);
  __syncthreads();
  mm64(lds + iY * 4096, lds + iT * 4096, lds + iF * 4096, 0, sqrtf(norm));
  __syncthreads();

  const float* R = lds + iF * 4096;
  float* sbuf = lds + iT * 4096;                   // dead buffer reused
  float* hbuf = sbuf + CDIM;
  if (tid < CDIM) {                                // s = mean over rows
    float ssum = 0.f;
    for (int i = 0; i < CDIM; ++i) ssum += R[i * CDIM + tid];
    sbuf[tid] = ssum * (1.0f / (float)CDIM);
  }
  __syncthreads();
  if (tid < CDIM / 8) {                            // h = relu(w1@s + b1), 8 wide
    float a = b1[tid];
    for (int c = 0; c < CDIM; ++c) a += w1[tid * CDIM + c] * sbuf[c];
    hbuf[tid] = a > 0.f ? a : 0.f;
  }
  __syncthreads();
  if (tid < CDIM) {                                // y = sigmoid(w2@h + b2)
    float a = b2[tid];
    for (int o = 0; o < CDIM / 8; ++o) a += w2[tid * 8 + o] * hbuf[o];
    yout[b * CDIM + tid] = 1.0f / (1.0f + expf(-a));
  }
}

// ---------------------------------------------------------------------------
// Pass 3: out = y[b][c] * (x - prev). Pure streaming, float4.
__global__ __launch_bounds__(256) void soca_scale(const float* __restrict__ prev,
                                                  const float* __restrict__ x,
                                                  const float* __restrict__ y,
                                                  float* __restrict__ out) {
  const size_t e = ((size_t)blockIdx.x * 256 + threadIdx.x) * 4;
  const float sc = y[e >> 14];                     // e / (H*W) == b*C + c
  const float4 xv = *(const float4*)(x + e);
  const float4 pv = *(const float4*)(prev + e);
  float4 o;
  o.x = (xv.x - pv.x) * sc; o.y = (xv.y - pv.y) * sc;
  o.z = (xv.z - pv.z) * sc; o.w = (xv.w - pv.w) * sc;
  *(float4*)(out + e) = o;
}

// ---------------------------------------------------------------------------
extern "C" void kernel_launch(void* const* d_in, const int* in_sizes, int n_in,
                              void* d_out, int out_size, void* d_ws, size_t ws_size,
                              hipStream_t stream) {
  const float* prev = (const float*)d_in[0];
  const float* x    = (const float*)d_in[1];
  const float* w1   = (const float*)d_in[2];
  const float* b1   = (const float*)d_in[3];
  const float* w2   = (const float*)d_in[4];
  const float* b2   = (const float*)d_in[5];
  float* out = (float*)d_out;

  float* gram   = (float*)d_ws;                    // 32*64*64
  float* rowsum = gram + BATCH * CDIM * CDIM;      // 32*64
  float* yv     = rowsum + BATCH * CDIM;           // 32*64
  const int nzero = BATCH * CDIM * CDIM + BATCH * CDIM;

  soca_zero<<<(nzero + 255) / 256, 256, 0, stream>>>(gram, nzero);
  soca_gram<<<BATCH * SPLITK, 256, 0, stream>>>(prev, x, gram, rowsum);
  soca_ns<<<BATCH, 256, 0, stream>>>(gram, rowsum, w1, b1, w2, b2, yv);
  const int nelem = BATCH * CDIM * 128 * 128;
  soca_scale<<<nelem / (256 * 4), 256, 0, stream>>>(prev, x, yv, out);
}